// TransformationModule_33397665694046
// MI455X (gfx1250) — compile-verified
//
#include <hip/hip_runtime.h>
#include <math.h>

#define FEAT 128
#define TOPK 6

typedef __attribute__((ext_vector_type(2))) float v2f;
typedef __attribute__((ext_vector_type(8))) float v8f;

// ---------------------------------------------------------------- utilities
__device__ inline float wred32(float v) {
  v += __shfl_xor(v, 1, 32);
  v += __shfl_xor(v, 2, 32);
  v += __shfl_xor(v, 4, 32);
  v += __shfl_xor(v, 8, 32);
  v += __shfl_xor(v, 16, 32);
  return v;
}

__device__ inline void topk_insert(float (&tv)[TOPK], int (&ti)[TOPK], float v, int i) {
  if (v > tv[TOPK - 1]) {
    tv[TOPK - 1] = v; ti[TOPK - 1] = i;
#pragma unroll
    for (int u = TOPK - 1; u > 0; --u) {
      if (tv[u] > tv[u - 1]) {
        float a = tv[u]; tv[u] = tv[u - 1]; tv[u - 1] = a;
        int   b = ti[u]; ti[u] = ti[u - 1]; ti[u - 1] = b;
      }
    }
  }
}

// ------------------------------------------------ kernel 1: ||t||^2 per row
__global__ void tnorm_kernel(const float* __restrict__ tf, float* __restrict__ tnorm, int T) {
  int t = blockIdx.x * blockDim.x + threadIdx.x;
  if (t >= T) return;
  const float4* p = (const float4*)(tf + (size_t)t * FEAT);
  float s = 0.f;
#pragma unroll
  for (int i = 0; i < FEAT / 4; ++i) {
    float4 v = p[i];
    s += v.x * v.x + v.y * v.y + v.z * v.z + v.w * v.w;
  }
  tnorm[t] = s;
}

// ---------------- kernel 2: fused fp32-WMMA GEMM + top-6 + softmax + gather
// 128 threads = 4 waves; each wave owns 16 source rows and streams all T
// target columns in 16-wide tiles through V_WMMA_F32_16X16X4_F32.
// B tiles are double-buffered in registers: tile t+1's 32 global_load_b64
// are issued before tile t's WMMA+scan work, hiding L2 latency.
__launch_bounds__(128)
__global__ void retrieve_kernel(const float* __restrict__ S,
                                const float* __restrict__ Tf,
                                const float* __restrict__ tn,
                                const float* __restrict__ tp,   // target_points [T,3]
                                float* __restrict__ pred,       // [N,3]
                                int N, int T) {
  __shared__ float sTile[4][16 * 17];
  __shared__ float sTn[4][16];
  __shared__ float sMV[4][16][TOPK];
  __shared__ int   sMI[4][16][TOPK];

  const int wave = threadIdx.x >> 5;
  const int lane = threadIdx.x & 31;
  const int half = lane >> 4;     // K-pair / column-half select
  const int l16  = lane & 15;

  const int r0   = (blockIdx.x * 4 + wave) * 16;
  const int rowA = min(r0 + l16, N - 1);

  // A block (16 rows x 128 feats) preloaded once: v2f per K-step per lane.
  // ISA 7.12.2 layout: lanes 0-15 carry K = kk,kk+1 ; lanes 16-31 K = kk+2,kk+3.
  v2f aReg[32];
  {
    const float* sp = S + (size_t)rowA * FEAT + 2 * half;
#pragma unroll
    for (int kk = 0; kk < 32; ++kk) aReg[kk] = *(const v2f*)(sp + kk * 4);
  }

  float tv[TOPK]; int ti[TOPK];
#pragma unroll
  for (int k = 0; k < TOPK; ++k) { tv[k] = -__builtin_inff(); ti[k] = 0; }

  const int numTiles = (T + 15) >> 4;

  v2f bA[32], bB[32];

  // batched B-tile load: 32 independent global_load_b64 in flight at once
  auto loadB = [&](int tile, v2f (&b)[32]) {
    const int cIdx = min((tile << 4) + l16, T - 1);
    const float* bp = Tf + (size_t)cIdx * FEAT + 2 * half;
#pragma unroll
    for (int kk = 0; kk < 32; ++kk) b[kk] = *(const v2f*)(bp + kk * 4);
  };

  // WMMA (two independent accumulator chains) + per-row top-6 scan
  auto computeTile = [&](int tile, const v2f (&b)[32]) {
    const int c0 = tile << 4;
    v8f acc0 = {}, acc1 = {};
#pragma unroll
    for (int kk = 0; kk < 16; ++kk)
      acc0 = __builtin_amdgcn_wmma_f32_16x16x4_f32(false, aReg[kk], false, b[kk],
                                                   (short)0, acc0, false, false);
#pragma unroll
    for (int kk = 16; kk < 32; ++kk)
      acc1 = __builtin_amdgcn_wmma_f32_16x16x4_f32(false, aReg[kk], false, b[kk],
                                                   (short)0, acc1, false, false);

    // C layout: VGPR r -> row r + 8*half, col l16   (wave-private LDS tile)
#pragma unroll
    for (int r = 0; r < 8; ++r)
      sTile[wave][(r + 8 * half) * 17 + l16] = acc0[r] + acc1[r];
    if (lane < 16) sTn[wave][lane] = tn[min(c0 + lane, T - 1)];
    __syncthreads();

    // lane owns row l16, column half `half` (8 cols) of this tile
#pragma unroll
    for (int q = 0; q < 8; ++q) {
      const int cl = 8 * half + q;
      const int gc = c0 + cl;
      if (gc < T) {
        float sVal = 2.0f * sTile[wave][l16 * 17 + cl] - sTn[wave][cl];
        topk_insert(tv, ti, sVal, gc);
      }
    }
  };

  // ping-pong pipeline (all conditions uniform across the block)
  loadB(0, bA);
  int tile = 0;
  while (true) {
    if (tile + 1 < numTiles) loadB(tile + 1, bB);
    computeTile(tile, bA);
    if (++tile >= numTiles) break;
    if (tile + 1 < numTiles) loadB(tile + 1, bA);
    computeTile(tile, bB);
    if (++tile >= numTiles) break;
  }

  // merge the two half-column top-6 lists per row
  __syncthreads();
  if (half) {
#pragma unroll
    for (int k = 0; k < TOPK; ++k) { sMV[wave][l16][k] = tv[k]; sMI[wave][l16][k] = ti[k]; }
  }
  __syncthreads();
  if (!half) {
#pragma unroll
    for (int k = 0; k < TOPK; ++k)
      topk_insert(tv, ti, sMV[wave][l16][k], sMI[wave][l16][k]);

    const int row = r0 + l16;
    if (row < N) {
      // softmax over top-6 (shift-invariant: equals reference softmax(-dists))
      float m = tv[0], w[TOPK], sum = 0.f;
#pragma unroll
      for (int k = 0; k < TOPK; ++k) { w[k] = expf(tv[k] - m); sum += w[k]; }
      float inv = 1.f / sum;
      float px = 0.f, py = 0.f, pz = 0.f;
#pragma unroll
      for (int k = 0; k < TOPK; ++k) {
        const float* q = tp + 3 * (size_t)ti[k];
        float wk = w[k] * inv;
        px += wk * q[0]; py += wk * q[1]; pz += wk * q[2];
      }
      pred[(size_t)row * 3 + 0] = px;
      pred[(size_t)row * 3 + 1] = py;
      pred[(size_t)row * 3 + 2] = pz;
    }
  }
}

// ---------------------------------------------- 3x3 Kabsch (all lanes redundant)
#define ROT_V(P, Q)                                                       \
  {                                                                       \
    float vp0 = v[0 * 3 + P], vp1 = v[1 * 3 + P], vp2 = v[2 * 3 + P];     \
    float vq0 = v[0 * 3 + Q], vq1 = v[1 * 3 + Q], vq2 = v[2 * 3 + Q];     \
    v[0 * 3 + P] = cc * vp0 - ss * vq0; v[1 * 3 + P] = cc * vp1 - ss * vq1; \
    v[2 * 3 + P] = cc * vp2 - ss * vq2;                                   \
    v[0 * 3 + Q] = ss * vp0 + cc * vq0; v[1 * 3 + Q] = ss * vp1 + cc * vq1; \
    v[2 * 3 + Q] = ss * vp2 + cc * vq2;                                   \
  }

__device__ inline void kabsch3(const float A[9], float R[9]) {
  // S = A^T A (symmetric)
  float s00 = A[0]*A[0] + A[3]*A[3] + A[6]*A[6];
  float s01 = A[0]*A[1] + A[3]*A[4] + A[6]*A[7];
  float s02 = A[0]*A[2] + A[3]*A[5] + A[6]*A[8];
  float s11 = A[1]*A[1] + A[4]*A[4] + A[7]*A[7];
  float s12 = A[1]*A[2] + A[4]*A[5] + A[7]*A[8];
  float s22 = A[2]*A[2] + A[5]*A[5] + A[8]*A[8];
  float v[9] = {1.f,0.f,0.f, 0.f,1.f,0.f, 0.f,0.f,1.f};

  for (int sweep = 0; sweep < 8; ++sweep) {
    if (fabsf(s01) > 1e-20f) {
      float tau = (s11 - s00) / (2.f * s01);
      float t = copysignf(1.f, tau) / (fabsf(tau) + sqrtf(1.f + tau * tau));
      float cc = rsqrtf(1.f + t * t), ss = t * cc;
      float n02 = cc * s02 - ss * s12, n12 = ss * s02 + cc * s12;
      s00 -= t * s01; s11 += t * s01; s01 = 0.f; s02 = n02; s12 = n12;
      ROT_V(0, 1);
    }
    if (fabsf(s02) > 1e-20f) {
      float tau = (s22 - s00) / (2.f * s02);
      float t = copysignf(1.f, tau) / (fabsf(tau) + sqrtf(1.f + tau * tau));
      float cc = rsqrtf(1.f + t * t), ss = t * cc;
      float n01 = cc * s01 - ss * s12, n12 = ss * s01 + cc * s12;
      s00 -= t * s02; s22 += t * s02; s02 = 0.f; s01 = n01; s12 = n12;
      ROT_V(0, 2);
    }
    if (fabsf(s12) > 1e-20f) {
      float tau = (s22 - s11) / (2.f * s12);
      float t = copysignf(1.f, tau) / (fabsf(tau) + sqrtf(1.f + tau * tau));
      float cc = rsqrtf(1.f + t * t), ss = t * cc;
      float n01 = cc * s01 - ss * s02, n02 = ss * s01 + cc * s02;
      s11 -= t * s12; s22 += t * s12; s12 = 0.f; s01 = n01; s02 = n02;
      ROT_V(1, 2);
    }
  }

  float w0 = s00, w1 = s11, w2 = s22;
  if (w0 < w1) { float t = w0; w0 = w1; w1 = t;
    t = v[0]; v[0] = v[1]; v[1] = t; t = v[3]; v[3] = v[4]; v[4] = t; t = v[6]; v[6] = v[7]; v[7] = t; }
  if (w0 < w2) { float t = w0; w0 = w2; w2 = t;
    t = v[0]; v[0] = v[2]; v[2] = t; t = v[3]; v[3] = v[5]; v[5] = t; t = v[6]; v[6] = v[8]; v[8] = t; }
  if (w1 < w2) { float t = w1; w1 = w2; w2 = t;
    t = v[1]; v[1] = v[2]; v[2] = t; t = v[4]; v[4] = v[5]; v[5] = t; t = v[7]; v[7] = v[8]; v[8] = t; }

  float x0 = v[0], x1 = v[3], x2 = v[6];   // v0 (largest sigma)
  float y0 = v[1], y1 = v[4], y2 = v[7];   // v1
  float z0 = x1 * y2 - x2 * y1;            // v2 = v0 x v1  -> det(V)=+1
  float z1 = x2 * y0 - x0 * y2;
  float z2 = x0 * y1 - x1 * y0;

  float u00 = A[0]*x0 + A[1]*x1 + A[2]*x2;
  float u01 = A[3]*x0 + A[4]*x1 + A[5]*x2;
  float u02 = A[6]*x0 + A[7]*x1 + A[8]*x2;
  float n0 = sqrtf(u00*u00 + u01*u01 + u02*u02);
  if (n0 > 1e-12f) { float inv = 1.f / n0; u00 *= inv; u01 *= inv; u02 *= inv; }
  else             { u00 = 1.f; u01 = 0.f; u02 = 0.f; }

  float u10 = A[0]*y0 + A[1]*y1 + A[2]*y2;
  float u11 = A[3]*y0 + A[4]*y1 + A[5]*y2;
  float u12 = A[6]*y0 + A[7]*y1 + A[8]*y2;
  float d01 = u00*u10 + u01*u11 + u02*u12;
  u10 -= d01 * u00; u11 -= d01 * u01; u12 -= d01 * u02;
  float n1 = sqrtf(u10*u10 + u11*u11 + u12*u12);
  if (n1 > 1e-12f) { float inv = 1.f / n1; u10 *= inv; u11 *= inv; u12 *= inv; }
  else {
    float ax = (fabsf(u00) < 0.9f) ? 1.f : 0.f, ay = 1.f - ax;
    u10 = -u02 * ay; u11 = u02 * ax; u12 = u00 * ay - u01 * ax;
    float inv = rsqrtf(u10*u10 + u11*u11 + u12*u12 + 1e-30f);
    u10 *= inv; u11 *= inv; u12 *= inv;
  }
  float u20 = u01 * u12 - u02 * u11;       // u2 = u0 x u1 -> det(U)=+1
  float u21 = u02 * u10 - u00 * u12;
  float u22 = u00 * u11 - u01 * u10;

  // R = U V^T  (equivalent to U diag(1,1,det) V^T of the true SVD)
  R[0] = u00*x0 + u10*y0 + u20*z0;  R[1] = u00*x1 + u10*y1 + u20*z1;  R[2] = u00*x2 + u10*y2 + u20*z2;
  R[3] = u01*x0 + u11*y0 + u21*z0;  R[4] = u01*x1 + u11*y1 + u21*z1;  R[5] = u01*x2 + u11*y2 + u21*z2;
  R[6] = u02*x0 + u12*y0 + u22*z0;  R[7] = u02*x1 + u12*y1 + u22*z1;  R[8] = u02*x2 + u12*y2 + u22*z2;
}

// --------------------- kernel 3: wave-per-node covariance + Procrustes (R, t)
__launch_bounds__(256)
__global__ void procrustes_kernel(const int* __restrict__ col,
                                  const float* __restrict__ pos,
                                  const float* __restrict__ pred,
                                  float* __restrict__ Rout,   // [N,9]
                                  float* __restrict__ tout,   // [N,3]
                                  int N, int deg) {
  const int wave = threadIdx.x >> 5;
  const int lane = threadIdx.x & 31;
  const int node = blockIdx.x * (blockDim.x >> 5) + wave;
  if (node >= N) return;

  float Sp0 = 0, Sp1 = 0, Sp2 = 0, Sq0 = 0, Sq1 = 0, Sq2 = 0;
  float M00 = 0, M01 = 0, M02 = 0, M10 = 0, M11 = 0, M12 = 0, M20 = 0, M21 = 0, M22 = 0;

  for (int j = lane; j < deg; j += 32) {
    int c = col[(size_t)node * deg + j];
    float p0 = pos[3 * (size_t)c + 0], p1 = pos[3 * (size_t)c + 1], p2 = pos[3 * (size_t)c + 2];
    float q0 = pred[3 * (size_t)c + 0], q1 = pred[3 * (size_t)c + 1], q2 = pred[3 * (size_t)c + 2];
    Sp0 += p0; Sp1 += p1; Sp2 += p2; Sq0 += q0; Sq1 += q1; Sq2 += q2;
    M00 += p0 * q0; M01 += p0 * q1; M02 += p0 * q2;
    M10 += p1 * q0; M11 += p1 * q1; M12 += p1 * q2;
    M20 += p2 * q0; M21 += p2 * q1; M22 += p2 * q2;
  }
  Sp0 = wred32(Sp0); Sp1 = wred32(Sp1); Sp2 = wred32(Sp2);
  Sq0 = wred32(Sq0); Sq1 = wred32(Sq1); Sq2 = wred32(Sq2);
  M00 = wred32(M00); M01 = wred32(M01); M02 = wred32(M02);
  M10 = wred32(M10); M11 = wred32(M11); M12 = wred32(M12);
  M20 = wred32(M20); M21 = wred32(M21); M22 = wred32(M22);

  float cnt = fmaxf((float)deg, 1.f), inv = 1.f / cnt;
  float cp0 = Sp0 * inv, cp1 = Sp1 * inv, cp2 = Sp2 * inv;
  float cq0 = Sq0 * inv, cq1 = Sq1 * inv, cq2 = Sq2 * inv;

  float Amat[9] = { M00 - Sp0 * Sq0 * inv, M01 - Sp0 * Sq1 * inv, M02 - Sp0 * Sq2 * inv,
                    M10 - Sp1 * Sq0 * inv, M11 - Sp1 * Sq1 * inv, M12 - Sp1 * Sq2 * inv,
                    M20 - Sp2 * Sq0 * inv, M21 - Sp2 * Sq1 * inv, M22 - Sp2 * Sq2 * inv };
  float R[9];
  kabsch3(Amat, R);

  float t0 = cq0 - (R[0] * cp0 + R[1] * cp1 + R[2] * cp2);
  float t1 = cq1 - (R[3] * cp0 + R[4] * cp1 + R[5] * cp2);
  float t2 = cq2 - (R[6] * cp0 + R[7] * cp1 + R[8] * cp2);

  if (lane == 0) {
#pragma unroll
    for (int k = 0; k < 9; ++k) Rout[(size_t)node * 9 + k] = R[k];
    tout[(size_t)node * 3 + 0] = t0;
    tout[(size_t)node * 3 + 1] = t1;
    tout[(size_t)node * 3 + 2] = t2;
  }
}

// ----------------------- kernel 4: residual under *neighbor* transforms (dst)
__launch_bounds__(256)
__global__ void residual_kernel(const int* __restrict__ col,
                                const float* __restrict__ pos,
                                const float* __restrict__ pred,
                                const float* __restrict__ Rout,
                                const float* __restrict__ tout,
                                float* __restrict__ dst,
                                int N, int deg) {
  const int wave = threadIdx.x >> 5;
  const int lane = threadIdx.x & 31;
  const int node = blockIdx.x * (blockDim.x >> 5) + wave;
  if (node >= N) return;

  float acc = 0.f;
  for (int j = lane; j < deg; j += 32) {
    int c = col[(size_t)node * deg + j];
    const float* R = Rout + (size_t)c * 9;
    const float* t = tout + (size_t)c * 3;
    float p0 = pos[3 * (size_t)c + 0], p1 = pos[3 * (size_t)c + 1], p2 = pos[3 * (size_t)c + 2];
    float q0 = pred[3 * (size_t)c + 0], q1 = pred[3 * (size_t)c + 1], q2 = pred[3 * (size_t)c + 2];
    float r0 = R[0] * p0 + R[1] * p1 + R[2] * p2 + t[0] - q0;
    float r1 = R[3] * p0 + R[4] * p1 + R[5] * p2 + t[1] - q1;
    float r2 = R[6] * p0 + R[7] * p1 + R[8] * p2 + t[2] - q2;
    acc += r0 * r0 + r1 * r1 + r2 * r2;
  }
  acc = wred32(acc);
  if (lane == 0) dst[node] = acc / fmaxf((float)deg, 1.f);
}

// --------------------------------------------------------------------- launch
extern "C" void kernel_launch(void* const* d_in, const int* in_sizes, int n_in,
                              void* d_out, int out_size, void* d_ws, size_t ws_size,
                              hipStream_t stream) {
  const float* S   = (const float*)d_in[0];   // source_feats [N,128]
  const float* Tf  = (const float*)d_in[1];   // target_feats [T,128]
  const float* tp  = (const float*)d_in[2];   // target_points [T,3]
  const float* pos = (const float*)d_in[3];   // pos [N,3]
  const int*   ei  = (const int*)d_in[4];     // edge_index [2,E]

  const int N   = in_sizes[0] / FEAT;
  const int T   = in_sizes[1] / FEAT;
  const int E   = in_sizes[4] / 2;
  const int deg = E / N;
  const int* col = ei + E;                    // edge_index[1]

  float* Rout = (float*)d_out;                // [N,9]
  float* tout = Rout + (size_t)N * 9;         // [N,3]
  float* dst  = tout + (size_t)N * 3;         // [N,1]

  float* tn   = (float*)d_ws;                                 // [T]
  size_t tnPad = (((size_t)T + 1023) / 1024) * 1024;
  float* pred = tn + tnPad;                                   // [N,3]

  tnorm_kernel<<<(T + 255) / 256, 256, 0, stream>>>(Tf, tn, T);

  const int blocksA = (N + 63) / 64;          // 4 waves * 16 rows per block
  retrieve_kernel<<<blocksA, 128, 0, stream>>>(S, Tf, tn, tp, pred, N, T);

  const int blocksB = (N + 7) / 8;            // 8 waves (nodes) per block
  procrustes_kernel<<<blocksB, 256, 0, stream>>>(col, pos, pred, Rout, tout, N, deg);
  residual_kernel<<<blocksB, 256, 0, stream>>>(col, pos, pred, Rout, tout, dst, N, deg);
}